// ContrastiveLoss_72043781423435
// MI455X (gfx1250) — compile-verified
//
#include <hip/hip_runtime.h>
#include <hip/hip_bf16.h>
#include <math.h>

// ---- problem constants (match reference) ----
#define Bn   8
#define Cc   64
#define Hh   32
#define Ww   32
#define HW   1024
#define NEG  256
#define TEMP 2.0f
#define FACTOR 0.8f
#define EPSV 1e-8f

typedef float v2f __attribute__((ext_vector_type(2)));
typedef float v8f __attribute__((ext_vector_type(8)));

// workspace layout (floats):
//   [0, 8192)        : n2[b][q]   = ||z2[b][:,q]||^2
//   [8192, 10240)    : Acc[b][n]  = sum_p sim_neg[p,n]
//   [10240, 10248)   : Acc0[b]    = sum_p sim0[p]
#define WS_N2   0
#define WS_ACC  8192
#define WS_ACC0 10240

// dynamic LDS layout (floats):
//   strip : 16*1024   (G strip for this pixel tile)
//   simg  : 3*1024    (img0 channels)
//   sn2   : 1024      (z2 column norms^2 for sample b)
//   sacc  : 256       (per-n partial sums)
#define SM_STRIP 0
#define SM_IMG   (16 * 1024)
#define SM_N2    (SM_IMG + 3 * 1024)
#define SM_ACC   (SM_N2 + 1024)
#define SM_FLOATS (SM_ACC + 256)
#define SM_BYTES (SM_FLOATS * 4)

// ------------------------------------------------------------------
// K1: per-sample z2 column squared norms + zero the accumulators
// ------------------------------------------------------------------
__global__ __launch_bounds__(256) void cl_prep_kernel(const float* __restrict__ z2,
                                                      float* __restrict__ ws) {
  int idx = blockIdx.x * 256 + threadIdx.x;   // 0 .. 8191
  if (idx < Bn * HW) {
    int b = idx >> 10;
    int q = idx & 1023;
    const float* z2b = z2 + (b << 16);        // b*64*1024
    float s = 0.f;
    #pragma unroll 8
    for (int c = 0; c < Cc; ++c) {
      float v = z2b[(c << 10) + q];
      s += v * v;
    }
    ws[WS_N2 + idx] = s;
  }
  if (idx < (Bn * NEG + Bn)) {                // zero Acc[8][256] and Acc0[8]
    ws[WS_ACC + idx] = 0.f;
  }
}

// ------------------------------------------------------------------
// K2: main kernel. One workgroup = (sample b, 16-pixel tile).
//   Phase 1: strip[16][1024] = z1_tile^T (16x64) x Z2_b (64x1024) via WMMA f32.
//   Phase 2: LDS gathers -> weights, cosine sims, per-n accumulation.
// ------------------------------------------------------------------
__global__ __launch_bounds__(128) void cl_main_kernel(const float* __restrict__ z1,
                                                      const float* __restrict__ z2,
                                                      const float* __restrict__ img,
                                                      const int*   __restrict__ nidx,
                                                      float* __restrict__ ws) {
  extern __shared__ float sm[];
  float* strip = sm + SM_STRIP;
  float* simg  = sm + SM_IMG;
  float* sn2   = sm + SM_N2;
  float* sacc  = sm + SM_ACC;

  const int wg   = blockIdx.x;        // 0 .. 511
  const int b    = wg >> 6;           // sample
  const int p0   = (wg & 63) << 4;    // tile base pixel
  const int tid  = threadIdx.x;
  const int lane = tid & 31;
  const int wave = tid >> 5;

  const float* z1b = z1 + (b << 16);
  const float* z2b = z2 + (b << 16);

  // stage img0 (3x1024), sample's n2 (1024), zero per-n accumulator
  for (int i = tid; i < 3 * HW; i += 128) simg[i] = img[i];
  for (int i = tid; i < HW; i += 128)     sn2[i]  = ws[WS_N2 + (b << 10) + i];
  for (int i = tid; i < NEG; i += 128)    sacc[i] = 0.f;

  // ---------------- Phase 1: WMMA strip ----------------
  // A fragment (16x4 f32, ISA layout): lanes 0-15 -> M=lane, K={0,1};
  // lanes 16-31 -> M=lane-16, K={2,3}.  16 K-chunks cover C=64.
  v2f afrag[16];
  {
    const int m  = lane & 15;
    const int ko = (lane >> 4) << 1;
    #pragma unroll
    for (int kk = 0; kk < 16; ++kk) {
      const int c = (kk << 2) + ko;
      afrag[kk].x = z1b[(c << 10) + p0 + m];
      afrag[kk].y = z1b[((c + 1) << 10) + p0 + m];
    }
  }
  {
    const int n  = lane & 15;
    const int ko = (lane >> 4) << 1;
    for (int t = 0; t < 16; ++t) {                  // 16 N-tiles per wave
      const int n0 = ((wave << 4) + t) << 4;        // tile column base
      v8f acc = {};
      #pragma unroll
      for (int kk = 0; kk < 16; ++kk) {
        // B fragment (4x16 f32): lanes 0-15 -> N=lane, K={0,1};
        // lanes 16-31 -> N=lane-16, K={2,3}
        const int c = (kk << 2) + ko;
        v2f bfrag;
        bfrag.x = z2b[(c << 10) + n0 + n];
        bfrag.y = z2b[((c + 1) << 10) + n0 + n];
        acc = __builtin_amdgcn_wmma_f32_16x16x4_f32(
            /*neg_a=*/false, afrag[kk], /*neg_b=*/false, bfrag,
            /*c_mod=*/(short)0, acc, /*reuse_a=*/false, /*reuse_b=*/false);
      }
      // D layout: VGPR j -> M=j (lanes 0-15) / M=j+8 (lanes 16-31), N=lane&15
      const int mbase = (lane < 16) ? 0 : 8;
      #pragma unroll
      for (int j = 0; j < 8; ++j)
        strip[((mbase + j) << 10) + n0 + n] = acc[j];
    }
  }
  __syncthreads();

  // ---------------- Phase 2: gathers + reductions ----------------
  const int pl = tid >> 3;             // local pixel 0..15
  const int g  = tid & 7;              // 8 threads per pixel, each owns 32 negs
  const int p  = p0 + pl;
  const float hp = (float)(p >> 5);
  const float wp = (float)(p & 31);

  const int* nr = nidx + (b << 19) + (p << 8);               // row indices
  const int* nc = nidx + (b << 19) + (1 << 18) + (p << 8);   // col indices

  const float c0 = simg[p];
  const float c1 = simg[HW + p];
  const float c2 = simg[2 * HW + p];

  float euc_acc = 0.f, rgb_acc = 0.f;
  int qs[32];
  #pragma unroll
  for (int k = 0; k < 32; ++k) {
    const int n  = g + (k << 3);
    const int r  = nr[n];
    const int w  = nc[n];
    const int q  = (r << 5) + w;
    qs[k] = q;
    const float dh = hp - (float)r;
    const float dw = wp - (float)w;
    euc_acc += dh * dh + dw * dw;
    const float d0 = c0 - simg[q];
    const float d1 = c1 - simg[HW + q];
    const float d2 = c2 - simg[2 * HW + q];
    rgb_acc += d0 * d0 + d1 * d1 + d2 * d2;
  }
  // z1 column squared norm, split over the 8-thread pixel group
  float zs = 0.f;
  #pragma unroll
  for (int c = g; c < Cc; c += 8) {
    const float v = z1b[(c << 10) + p];
    zs += v * v;
  }
  // reduce over the 8-lane pixel group (all within one wave32)
  for (int off = 4; off; off >>= 1) {
    euc_acc += __shfl_xor(euc_acc, off, 8);
    rgb_acc += __shfl_xor(rgb_acc, off, 8);
    zs      += __shfl_xor(zs, off, 8);
  }
  const float euc_norm = sqrtf((float)((Hh - 1) * (Hh - 1) + (Ww - 1) * (Ww - 1)));
  const float weight = sqrtf(euc_acc) / euc_norm * FACTOR +
                       sqrtf(rgb_acc) / sqrtf(3.0f) * (1.0f - FACTOR);
  const float z1n = sqrtf(zs);

  if (g == 0) {
    const float cos0 = zs / fmaxf(zs, EPSV);
    atomicAdd(&ws[WS_ACC0 + b], fminf(fabsf(cos0), 1.0f));
  }

  #pragma unroll
  for (int k = 0; k < 32; ++k) {
    const int n = g + (k << 3);
    const int q = qs[k];
    const float dot   = strip[(pl << 10) + q];
    const float denom = fmaxf(z1n * sqrtf(sn2[q]), EPSV);
    const float s     = fminf(fabsf((dot / denom) * weight), 1.0f);
    atomicAdd(&sacc[n], s);          // ds_add_f32
  }
  __syncthreads();
  for (int i = tid; i < NEG; i += 128)
    atomicAdd(&ws[WS_ACC + (b << 8) + i], sacc[i]);
}

// ------------------------------------------------------------------
// K3: finalize -> 3 scalars
// ------------------------------------------------------------------
__global__ __launch_bounds__(256) void cl_fin_kernel(const float* __restrict__ ws,
                                                     float* __restrict__ out) {
  __shared__ float red_t[256];
  __shared__ float red_s[256];
  const int n = threadIdx.x;

  float term_sum = 0.f;   // sum over b of max(log1p(-T/TEMP), -100) for this n
  float t_sum    = 0.f;   // sum over b of T_b[n]
  for (int b = 0; b < Bn; ++b) {
    const float T    = ws[WS_ACC + (b << 8) + n] * (1.0f / (float)HW);
    const float simn = T / TEMP;
    term_sum += fmaxf(log1pf(-simn), -100.0f);
    t_sum    += T;
  }
  red_t[n] = term_sum;
  red_s[n] = t_sum;
  __syncthreads();
  for (int off = 128; off; off >>= 1) {
    if (n < off) {
      red_t[n] += red_t[n + off];
      red_s[n] += red_s[n + off];
    }
    __syncthreads();
  }
  if (n == 0) {
    float lp_sum = 0.f, s0_sum = 0.f;
    for (int b = 0; b < Bn; ++b) {
      const float S0 = ws[WS_ACC0 + b] * (1.0f / (float)HW);
      lp_sum += fmaxf(logf(S0), -100.0f);
      s0_sum += S0;
    }
    // loss = mean_b [ -(logp + sum log1m) / (NEG+1) ]
    out[0] = -(lp_sum + red_t[0]) / ((float)Bn * (float)(NEG + 1));
    out[1] = s0_sum / (float)Bn;
    // sum(sim_all[1:]) / NEG * TEMP / B  with sim[1+n] = T/TEMP
    out[2] = red_s[0] / ((float)NEG * (float)Bn);
  }
}

// ------------------------------------------------------------------
extern "C" void kernel_launch(void* const* d_in, const int* in_sizes, int n_in,
                              void* d_out, int out_size, void* d_ws, size_t ws_size,
                              hipStream_t stream) {
  (void)in_sizes; (void)n_in; (void)out_size; (void)ws_size;
  const float* views_1 = (const float*)d_in[0];
  const float* views_2 = (const float*)d_in[1];
  const float* img     = (const float*)d_in[2];
  const int*   neg_idx = (const int*)d_in[3];
  float* out = (float*)d_out;
  float* ws  = (float*)d_ws;

  cl_prep_kernel<<<(Bn * HW + 255) / 256, 256, 0, stream>>>(views_2, ws);
  cl_main_kernel<<<Bn * (HW / 16), 128, SM_BYTES, stream>>>(views_1, views_2, img,
                                                            neg_idx, ws);
  cl_fin_kernel<<<1, 256, 0, stream>>>(ws, out);
}